// EMAValueMixer_3049426780599
// MI455X (gfx1250) — compile-verified
//
#include <hip/hip_runtime.h>
#include <math.h>

// EMA scan: out[b,t,d] = a_d*v[b,t,d] + (1-a_d)*out[b,t-1,d], out[-1]=0
// 3-phase chunked linear recurrence; phase 3 uses V_WMMA_F32_16X16X4_F32
// to compute 16-step prefix scans as D = L @ Y (L = lower-triangular ones).

typedef __attribute__((ext_vector_type(2))) float v2f;
typedef __attribute__((ext_vector_type(8))) float v8f;

constexpr int Bc  = 8;
constexpr int Tc  = 4096;
constexpr int Dc  = 1024;
constexpr int NCH = 32;        // time chunks
constexpr int CL  = Tc / NCH;  // 128 timesteps per chunk
constexpr int NDG = Dc / 16;   // 64 channel groups of 16

__device__ __forceinline__ float sigmoidf_fast(float x) {
    return 1.0f / (1.0f + __expf(-x));
}

// ---------------- Phase 1: chunk-local EMA end values (zero init) ---------
// thread (b, c, d): scan 128 steps, keep only final value. Wave-coalesced
// 128B loads per step; pulls all of v into L2 (134MB < 192MB).
__global__ __launch_bounds__(256) void ema_phase1(
    const float* __restrict__ v, const float* __restrict__ logit,
    float* __restrict__ ends) {
    int g = blockIdx.x * 256 + threadIdx.x;     // B*NCH*Dc threads
    int d = g & (Dc - 1);
    int c = (g >> 10) & (NCH - 1);
    int b = g >> 15;
    float a = sigmoidf_fast(logit[d]);
    float w = 1.0f - a;
    const float* vp = v + ((size_t)(b * Tc + c * CL)) * Dc + d;
    float cur = 0.0f;
#pragma unroll 4
    for (int t = 0; t < CL; ++t)
        cur = fmaf(w, cur, a * vp[(size_t)t * Dc]);
    ends[(b * NCH + c) * Dc + d] = cur;
}

// ---------------- Phase 2: serial carry scan over chunks ------------------
// carry[c] = EMA state entering chunk c. carry[0]=0;
// carry[c] = ends[c-1] + w^CL * carry[c-1].
__global__ __launch_bounds__(256) void ema_phase2(
    const float* __restrict__ ends, const float* __restrict__ logit,
    float* __restrict__ carry) {
    int g = blockIdx.x * 256 + threadIdx.x;     // B*Dc threads
    int d = g & (Dc - 1);
    int b = g >> 10;
    float a = sigmoidf_fast(logit[d]);
    float w = 1.0f - a;
    float wL = w;
#pragma unroll
    for (int i = 0; i < 7; ++i) wL *= wL;       // w^128 (CL = 2^7)
    float P = 0.0f;
    for (int c = 0; c < NCH; ++c) {
        carry[(b * NCH + c) * Dc + d] = P;
        P = fmaf(wL, P, ends[(b * NCH + c) * Dc + d]);
    }
}

// ---------------- Phase 3: WMMA prefix-scan rescan with exact carries -----
// Per 16x16 tile (i = time row, d = channel col), with state P = out[t0-1]:
//   Y[j,d]   = w_d^{-j} * a_d * v[t0+j, d]
//   psum     = L @ Y            (L = 16x16 lower-triangular ones, via
//                                4 chained V_WMMA_F32_16X16X4_F32, K=4 each)
//   out[t0+i,d] = w_d^i * psum[i,d] + w_d^{i+1} * P
// f32 operand layout (ISA 7.12.2): lanes 0-15 hold K = {4k,4k+1} in
// (v0,v1) of slice k; lanes 16-31 hold K = {4k+2,4k+3}. C/D: vgpr r holds
// row M = r + 8*(lane>=16), col N = lane&15.
__global__ __launch_bounds__(256) void ema_phase3(
    const float* __restrict__ v, const float* __restrict__ logit,
    const float* __restrict__ carry, float* __restrict__ out) {
    const int lane = threadIdx.x & 31;
    const int wave = blockIdx.x * 8 + (threadIdx.x >> 5);
    const int dg = wave & (NDG - 1);
    const int c  = (wave >> 6) & (NCH - 1);
    const int b  = wave >> 11;
    const int dl = lane & 15;                  // channel within group = N = M
    const int d  = dg * 16 + dl;
    const bool hi = lane >= 16;
    const int jb = hi ? 2 : 0;                 // K offset of this lane half
    const int ib = hi ? 8 : 0;                 // output-row offset (C/D layout)

    float a  = sigmoidf_fast(logit[d]);
    float w  = 1.0f - a;
    float wi = 1.0f / w;                       // w^-1
    float wi2 = wi * wi;
    float wi4 = wi2 * wi2;                     // w^-4 (per-k step)
    const float wq0_init = hi ? wi2 : 1.0f;    // w^-jb
    float wbase = 1.0f;                        // w^ib
    if (hi) { float w2 = w * w, w4 = w2 * w2; wbase = w4 * w4; }

    float P = carry[(b * NCH + c) * Dc + d];   // EMA state entering chunk
    const size_t vb = (size_t)b * Tc * Dc;
    const float* vp = v + vb + d;
    float* op = out + vb + d;

    for (int tau = 0; tau < CL / 16; ++tau) {
        const int t0 = c * CL + tau * 16;
        v8f acc = {0.f, 0.f, 0.f, 0.f, 0.f, 0.f, 0.f, 0.f};
        float wq0 = wq0_init;                  // w^-(4k+jb)
#pragma unroll
        for (int k = 0; k < 4; ++k) {
            const int j0 = 4 * k + jb;
            float x0 = vp[(size_t)(t0 + j0) * Dc];
            float x1 = vp[(size_t)(t0 + j0 + 1) * Dc];
            v2f bm;                            // B = Y rows j0, j0+1
            bm.x = wq0 * a * x0;
            bm.y = (wq0 * wi) * a * x1;
            v2f am;                            // A = L[:, j0], L[:, j0+1]
            am.x = (j0     <= dl) ? 1.0f : 0.0f;
            am.y = (j0 + 1 <= dl) ? 1.0f : 0.0f;
            acc = __builtin_amdgcn_wmma_f32_16x16x4_f32(
                      false, am, false, bm, (short)0, acc, false, false);
            wq0 *= wi4;
        }
        // unscale + apply carry: out_i = w^i * (psum_i + w*P)
        const float wpP = w * P;
        float wr = wbase;                      // w^(ib+r)
        float olast = 0.0f;
        #pragma unroll
        for (int r = 0; r < 8; ++r) {
            float o = wr * (acc[r] + wpP);
            // NT store: don't let the 134MB output stream evict v from L2
            __builtin_nontemporal_store(o, &op[(size_t)(t0 + ib + r) * Dc]);
            olast = o;
            wr *= w;
        }
        // next tile's state = out row i=15 -> lanes 16-31, r=7; broadcast
        // to both lanes of each channel
        P = __shfl(olast, 16 + dl, 32);
    }
}

extern "C" void kernel_launch(void* const* d_in, const int* in_sizes, int n_in,
                              void* d_out, int out_size, void* d_ws, size_t ws_size,
                              hipStream_t stream) {
    const float* v     = (const float*)d_in[0];
    const float* logit = (const float*)d_in[1];
    float* out = (float*)d_out;
    // workspace: ends + carries, each B*NCH*Dc floats (1 MB each)
    float* ends  = (float*)d_ws;
    float* carry = ends + (size_t)Bc * NCH * Dc;

    ema_phase1<<<(Bc * NCH * Dc) / 256, 256, 0, stream>>>(v, logit, ends);
    ema_phase2<<<(Bc * Dc) / 256, 256, 0, stream>>>(ends, logit, carry);
    ema_phase3<<<(Bc * NCH * NDG) / 8, 256, 0, stream>>>(v, logit, carry, out);
}